// Seq2Seq_66632122630152
// MI455X (gfx1250) — compile-verified
//
#include <hip/hip_runtime.h>
#include <hip/hip_bf16.h>
#include <math.h>

// ---------------------------------------------------------------------------
// Seq2Seq (bi-LSTM encoder + 1-step attention decoder) for gfx1250 / MI455X.
// - All dense math on the fp32 matrix pipe: V_WMMA_F32_16X16X4_F32.
// - b128 operand loads feed TWO WMMAs each (K-permutation consistent on A/B).
// - Recurrence: ONE persistent kernel, 32 blocks, per-step grid barrier;
//   each block keeps its 256KB Whh slice resident in CDNA5's 320KB LDS,
//   staged via async global->LDS loads (ASYNCcnt).
// - Streaming GEMMs prefetch the weight stream (global_prefetch_b8).
// ---------------------------------------------------------------------------

typedef __attribute__((ext_vector_type(2))) float v2f;
typedef __attribute__((ext_vector_type(4))) float v4f;
typedef __attribute__((ext_vector_type(8))) float v8f;
typedef __attribute__((ext_vector_type(4))) int   v4i;

#define B_  16
#define L_  128
#define D_  512
#define E_  512
#define G4  2048   // 4*D
#define DI_ 2048   // E + E + 2D

// Async global->LDS copy path (guarded: falls back to plain copies).
#if defined(__has_builtin)
#if __has_builtin(__builtin_amdgcn_global_load_async_to_lds_b128) && \
    __has_builtin(__builtin_amdgcn_s_wait_asynccnt)
#define ASYNC_LDS 1
#endif
#endif
#ifndef ASYNC_LDS
#define ASYNC_LDS 0
#endif

static __device__ __forceinline__ float sigf(float x) {
  return 1.0f / (1.0f + __expf(-x));
}

static __device__ __forceinline__ v8f wmma4(v2f a, v2f b, v8f c) {
  // D = A(16x4 f32) * B(4x16 f32) + C(16x16 f32)
  return __builtin_amdgcn_wmma_f32_16x16x4_f32(
      /*neg_a=*/false, a, /*neg_b=*/false, b,
      /*c_mod=*/(short)0, c, /*reuse_a=*/false, /*reuse_b=*/false);
}

// ---------------------------------------------------------------------------
// Generic fp32 WMMA GEMM:  C[M x N] = act( A[M x K] @ W[N x K]^T + bias + Cin )
// Grid: (M/16, N/256). Block: 256 threads = 8 waves; each wave computes TWO
// adjacent 16x16 tiles. Per 8-K group: one b128 load per operand, split into
// two v2f sub-fragments (K sets {k,k+1,k+4,k+5} / {k+2,k+3,k+6,k+7} — the
// same permutation on A and B, so the contraction is exact). Four independent
// WMMA accumulator chains. act: 0 = none, 1 = tanh. Craw: pre-act copy.
// ---------------------------------------------------------------------------
__global__ __launch_bounds__(256) void gemm_wmma_kernel(
    const float* __restrict__ A, int lda,
    const float* __restrict__ W, int ldw,
    const float* __restrict__ bias,
    const float* Cin,        // may alias C (accumulate in-place)
    float* C, int ldc,
    float* Craw,
    int K, int act)
{
  const int lane = threadIdx.x & 31;
  const int wave = threadIdx.x >> 5;
  const int half = lane >> 4;      // 0: lanes 0-15, 1: lanes 16-31
  const int l15  = lane & 15;
  const int mrow0 = blockIdx.x * 16;
  const int ncol0 = (blockIdx.y * 16 + wave * 2) * 16;

  // b128 per lane at k + 4*half: half0 covers K {k..k+3}, half1 {k+4..k+7}
  const float* Ap  = A + (size_t)(mrow0 + l15) * lda + 4 * half;
  const float* Wp0 = W + (size_t)(ncol0 + l15) * ldw + 4 * half;
  const float* Wp1 = W + (size_t)(ncol0 + 16 + l15) * ldw + 4 * half;

  v8f acc0a = {}, acc0b = {}, acc1a = {}, acc1b = {};
  for (int kk = 0; kk < K; kk += 64) {
    // prefetch the streaming weight operand ahead (global_prefetch_b8)
    __builtin_prefetch(Wp0 + kk + 512, 0, 1);
    __builtin_prefetch(Wp1 + kk + 512, 0, 1);
#pragma unroll
    for (int k = kk; k < kk + 64; k += 8) {
      v4f a  = *(const v4f*)(Ap + k);
      v4f b0 = *(const v4f*)(Wp0 + k);
      v4f b1 = *(const v4f*)(Wp1 + k);
      v2f a01 = {a.x, a.y},  a23 = {a.z, a.w};
      v2f b0a = {b0.x, b0.y}, b0b = {b0.z, b0.w};
      v2f b1a = {b1.x, b1.y}, b1b = {b1.z, b1.w};
      acc0a = wmma4(a01, b0a, acc0a);
      acc0b = wmma4(a23, b0b, acc0b);
      acc1a = wmma4(a01, b1a, acc1a);
      acc1b = wmma4(a23, b1b, acc1b);
    }
  }

  const int col0 = ncol0 + l15;
  const int col1 = col0 + 16;
  const float bv0 = bias ? bias[col0] : 0.0f;
  const float bv1 = bias ? bias[col1] : 0.0f;
#pragma unroll
  for (int r = 0; r < 8; ++r) {
    const int row = mrow0 + r + 8 * half;   // C/D: M = r + 8*half
    float v0 = acc0a[r] + acc0b[r] + bv0;
    float v1 = acc1a[r] + acc1b[r] + bv1;
    if (Cin) {
      v0 += Cin[(size_t)row * ldc + col0];
      v1 += Cin[(size_t)row * ldc + col1];
    }
    if (Craw) {
      Craw[(size_t)row * ldc + col0] = v0;
      Craw[(size_t)row * ldc + col1] = v1;
    }
    C[(size_t)row * ldc + col0] = (act == 1) ? tanhf(v0) : v0;
    C[(size_t)row * ldc + col1] = (act == 1) ? tanhf(v1) : v1;
  }
}

// ---------------------------------------------------------------------------
// Embedding gather: emb[b*L+t][:] = word_emb[x_train[b*L+t]][:]
// ---------------------------------------------------------------------------
__global__ __launch_bounds__(128) void embed_gather_kernel(
    const int* __restrict__ x_train, const float* __restrict__ word_emb,
    float* __restrict__ emb)
{
  const int row = blockIdx.x;          // b*L + t
  const int tok = x_train[row];
  const float4* src = (const float4*)(word_emb + (size_t)tok * E_);
  float4* dst = (float4*)(emb + (size_t)row * E_);
  dst[threadIdx.x] = src[threadIdx.x]; // 128 threads * 16B = 2KB row
}

// ---------------------------------------------------------------------------
// Init: zero h double-buffers and arrival counters (must run every call).
// ---------------------------------------------------------------------------
__global__ void init_kernel(float* hbufs, int* cnt) {
  const int i = blockIdx.x * blockDim.x + threadIdx.x;
  if (i < 2 * 2 * B_ * D_) hbufs[i] = 0.0f;
  if (i < 2 * L_) cnt[i] = 0;
}

// ---------------------------------------------------------------------------
// Persistent bidirectional LSTM scan.
// Grid = 32 blocks: dir = bx>>4 (0 fwd / 1 bwd), 16 blocks per direction,
// each owns hidden slice j in [blk*32, blk*32+32).
// The block's Whh slice (128 gate-cols x 512 K = 256KB) is staged into LDS
// ONCE (async global->LDS) and feeds the WMMA B operand via ds_load_b128 for
// all 128 steps; the A operand (h state, 32KB) stays L2-hot.
// Per-direction per-step arrival counters form the cross-WGP grid barrier.
// ---------------------------------------------------------------------------
#define SCAN_SMEM_FLOATS (128 * D_ + 4 * B_ * 32)
#define SCAN_SMEM_BYTES  (SCAN_SMEM_FLOATS * 4)

__global__ __launch_bounds__(256) void lstm_scan_kernel(
    const float* __restrict__ xg_f, const float* __restrict__ xg_b,
    const float* __restrict__ Whh_f, const float* __restrict__ Whh_b,
    const float* __restrict__ bhh_f, const float* __restrict__ bhh_b,
    float* hbufs,                // [dir][2][B_*D_] double buffer
    float* __restrict__ x_enc,   // [B_*L_][2*D_]
    float* __restrict__ cT_cat,  // [B_][2*D_]
    int* cnt)                    // [2][L_]
{
  extern __shared__ float smem[];
  float* sh_w = smem;              // [128][512]  Whh slice   (256 KB)
  float* gbuf = smem + 128 * D_;   // [4][16][32] gate tile   (8 KB)

  const int dir  = blockIdx.x >> 4;
  const int blk  = blockIdx.x & 15;
  const int j0   = blk * 32;
  const int tid  = threadIdx.x;
  const int lane = tid & 31;
  const int wave = tid >> 5;
  const int gate = wave >> 1;          // 0:i 1:f 2:g 3:o
  const int sub  = wave & 1;
  const int half = lane >> 4;
  const int l15  = lane & 15;
  const int gcol = gate * D_ + j0 + sub * 16 + l15;

  const float* xg  = dir ? xg_b  : xg_f;
  const float* Whh = dir ? Whh_b : Whh_f;
  const float* bhh = dir ? bhh_b : bhh_f;
  float* hb   = hbufs + (size_t)dir * 2 * B_ * D_;
  int* mycnt  = cnt + dir * L_;
  const float bhh_v = bhh[gcol];

  // ---- stage this block's Whh slice into LDS (once, reused 128 steps) ----
  for (int c = wave; c < 128; c += 8) {
    const int gc = (c >> 5) * D_ + j0 + (c & 31);
    const float* src = Whh + (size_t)gc * D_;
    float* dst = sh_w + c * D_;
#if ASYNC_LDS
    for (int k4 = lane; k4 < D_ / 4; k4 += 32)
      __builtin_amdgcn_global_load_async_to_lds_b128(
          (v4i*)(void*)(src + 4 * k4), (v4i*)(void*)(dst + 4 * k4), 0, 0);
#else
    for (int k4 = lane; k4 < D_ / 4; k4 += 32)
      *(float4*)(dst + 4 * k4) = *(const float4*)(src + 4 * k4);
#endif
  }
#if ASYNC_LDS
  __builtin_amdgcn_s_wait_asynccnt(0);
#endif
  __syncthreads();

  // B-operand pointer into LDS for this wave's 16 gate columns
  const float* Bp = sh_w + (size_t)(wave * 16 + l15) * D_ + 4 * half;

  // Each thread owns 2 (batch, j-local) cell-state elements, in registers.
  const int p0  = tid * 2;
  const int cb0 = p0 >> 5,       cj0 = p0 & 31;
  const int cb1 = (p0 + 1) >> 5, cj1 = (p0 + 1) & 31;
  float cr0 = 0.0f, cr1 = 0.0f;

  for (int s = 0; s < L_; ++s) {
    const int t = dir ? (L_ - 1 - s) : s;
    const float* Ap = hb + (s & 1) * (B_ * D_) + (size_t)l15 * D_ + 4 * half;

    v8f acc0 = {}, acc1 = {};     // two independent WMMA chains over K
#pragma unroll 8
    for (int k = 0; k < D_; k += 8) {
      v4f a = *(const v4f*)(Ap + k);   // global_load_b128 (L2-hot h state)
      v4f b = *(const v4f*)(Bp + k);   // ds_load_b128 (LDS-resident Whh)
      v2f a01 = {a.x, a.y}, a23 = {a.z, a.w};
      v2f b01 = {b.x, b.y}, b23 = {b.z, b.w};
      acc0 = wmma4(a01, b01, acc0);
      acc1 = wmma4(a23, b23, acc1);
    }

#pragma unroll
    for (int r = 0; r < 8; ++r) {
      const int brow = r + 8 * half;   // batch row
      const float pre = acc0[r] + acc1[r] + bhh_v +
                        xg[(size_t)(brow * L_ + t) * G4 + gcol];
      gbuf[(gate * B_ + brow) * 32 + sub * 16 + l15] = pre;
    }
    __syncthreads();

    {
      const float iv0 = gbuf[(0 * B_ + cb0) * 32 + cj0];
      const float fv0 = gbuf[(1 * B_ + cb0) * 32 + cj0];
      const float gv0 = gbuf[(2 * B_ + cb0) * 32 + cj0];
      const float ov0 = gbuf[(3 * B_ + cb0) * 32 + cj0];
      cr0 = sigf(fv0) * cr0 + sigf(iv0) * tanhf(gv0);
      const float hv0 = sigf(ov0) * tanhf(cr0);

      const float iv1 = gbuf[(0 * B_ + cb1) * 32 + cj1];
      const float fv1 = gbuf[(1 * B_ + cb1) * 32 + cj1];
      const float gv1 = gbuf[(2 * B_ + cb1) * 32 + cj1];
      const float ov1 = gbuf[(3 * B_ + cb1) * 32 + cj1];
      cr1 = sigf(fv1) * cr1 + sigf(iv1) * tanhf(gv1);
      const float hv1 = sigf(ov1) * tanhf(cr1);

      float* Hn = hb + ((s + 1) & 1) * (B_ * D_);
      Hn[cb0 * D_ + j0 + cj0] = hv0;
      Hn[cb1 * D_ + j0 + cj1] = hv1;
      x_enc[(size_t)(cb0 * L_ + t) * (2 * D_) + dir * D_ + j0 + cj0] = hv0;
      x_enc[(size_t)(cb1 * L_ + t) * (2 * D_) + dir * D_ + j0 + cj1] = hv1;
      if (s == L_ - 1) {
        cT_cat[cb0 * (2 * D_) + dir * D_ + j0 + cj0] = cr0;
        cT_cat[cb1 * (2 * D_) + dir * D_ + j0 + cj1] = cr1;
      }
    }
    __syncthreads();

    // grid barrier across the 16 blocks of this direction
    __threadfence();
    if (tid == 0)
      __hip_atomic_fetch_add(&mycnt[s], 1, __ATOMIC_RELEASE, __HIP_MEMORY_SCOPE_AGENT);
    while (__hip_atomic_load(&mycnt[s], __ATOMIC_ACQUIRE, __HIP_MEMORY_SCOPE_AGENT) < 16)
      __builtin_amdgcn_s_sleep(1);
    __threadfence();
    __syncthreads();
  }
}

// ---------------------------------------------------------------------------
// Decoder input build: xin = [word_emb[x_train[:,0]], attr, zeros(2D)]
// ---------------------------------------------------------------------------
__global__ void dec_xin_kernel(const int* __restrict__ x_train,
                               const int* __restrict__ y_train,
                               const float* __restrict__ word_emb,
                               const float* __restrict__ attr_emb,
                               float* __restrict__ xin)
{
  const int b   = blockIdx.x;
  const int tok = x_train[b * L_ + 0];
  const int y0  = y_train[b * 2 + 0];
  const int y1  = y_train[b * 2 + 1];
  for (int e = threadIdx.x; e < E_; e += blockDim.x) {
    xin[b * DI_ + e]      = word_emb[(size_t)tok * E_ + e];
    xin[b * DI_ + E_ + e] = attr_emb[y0 * E_ + e] + attr_emb[y1 * E_ + e];
  }
  for (int e = threadIdx.x; e < 2 * D_; e += blockDim.x)
    xin[b * DI_ + 2 * E_ + e] = 0.0f;      // feed0
}

// ---------------------------------------------------------------------------
// Decoder LSTM gate nonlinearity (single step).
// ---------------------------------------------------------------------------
__global__ void dec_gates_kernel(const float* __restrict__ g,
                                 const float* __restrict__ c0,
                                 float* __restrict__ h1, float* __restrict__ c1)
{
  const int idx = blockIdx.x * blockDim.x + threadIdx.x;  // b*512 + j
  const int b = idx >> 9, j = idx & 511;
  const float iv = g[b * G4 + j];
  const float fv = g[b * G4 + D_ + j];
  const float gv = g[b * G4 + 2 * D_ + j];
  const float ov = g[b * G4 + 3 * D_ + j];
  const float c  = sigf(fv) * c0[idx] + sigf(iv) * tanhf(gv);
  c1[idx] = c;
  h1[idx] = sigf(ov) * tanhf(c);
}

// ---------------------------------------------------------------------------
// Attention: per-batch softmax over L=128 positions + context vector.
// ---------------------------------------------------------------------------
__global__ __launch_bounds__(128) void attention_kernel(
    const float* __restrict__ x_enc_k, const float* __restrict__ x_enc,
    const float* __restrict__ tq, const float* __restrict__ w_att,
    const float* __restrict__ b_att, float* __restrict__ ctx)
{
  const int b = blockIdx.x;
  const int l = threadIdx.x;            // 0..127
  __shared__ float sw[L_];
  __shared__ float red[L_];

  float s = b_att[0];
  const float* xk  = x_enc_k + (size_t)(b * L_ + l) * D_;
  const float* tqb = tq + b * D_;
  for (int d = 0; d < D_; ++d)
    s += tanhf(xk[d] + tqb[d]) * w_att[d];

  red[l] = s; __syncthreads();
  for (int off = 64; off; off >>= 1) {
    if (l < off) red[l] = fmaxf(red[l], red[l + off]);
    __syncthreads();
  }
  const float m = red[0]; __syncthreads();
  const float e = __expf(s - m);
  red[l] = e; __syncthreads();
  for (int off = 64; off; off >>= 1) {
    if (l < off) red[l] += red[l + off];
    __syncthreads();
  }
  sw[l] = e / red[0];
  __syncthreads();

  for (int d = l; d < 2 * D_; d += 128) {
    float a = 0.0f;
    for (int ll = 0; ll < L_; ++ll)
      a += sw[ll] * x_enc[(size_t)(b * L_ + ll) * (2 * D_) + d];
    ctx[b * (2 * D_) + d] = a;
  }
}

// ---------------------------------------------------------------------------
// hc = concat(h1, ctx)  -> (16, 1536)
// ---------------------------------------------------------------------------
__global__ void hc_build_kernel(const float* __restrict__ h1,
                                const float* __restrict__ ctx,
                                float* __restrict__ hc)
{
  const int b = blockIdx.x;
  for (int d = threadIdx.x; d < D_; d += blockDim.x)
    hc[b * (3 * D_) + d] = h1[b * D_ + d];
  for (int d = threadIdx.x; d < 2 * D_; d += blockDim.x)
    hc[b * (3 * D_) + D_ + d] = ctx[b * (2 * D_) + d];
}

// ---------------------------------------------------------------------------
extern "C" void kernel_launch(void* const* d_in, const int* in_sizes, int n_in,
                              void* d_out, int out_size, void* d_ws, size_t ws_size,
                              hipStream_t stream)
{
  const int*   x_train  = (const int*)d_in[0];
  const int*   y_train  = (const int*)d_in[2];
  const float* word_emb = (const float*)d_in[3];
  const float* attr_emb = (const float*)d_in[4];
  const float* Wih_f  = (const float*)d_in[5];
  const float* Whh_f  = (const float*)d_in[6];
  const float* bih_f  = (const float*)d_in[7];
  const float* bhh_f  = (const float*)d_in[8];
  const float* Wih_b  = (const float*)d_in[9];
  const float* Whh_b  = (const float*)d_in[10];
  const float* bih_b  = (const float*)d_in[11];
  const float* bhh_b  = (const float*)d_in[12];
  const float* W_bridge = (const float*)d_in[13];
  const float* W_enc2k  = (const float*)d_in[14];
  const float* Wih_d  = (const float*)d_in[15];
  const float* Whh_d  = (const float*)d_in[16];
  const float* bih_d  = (const float*)d_in[17];
  const float* bhh_d  = (const float*)d_in[18];
  const float* W_trg  = (const float*)d_in[19];
  const float* b_trg  = (const float*)d_in[20];
  const float* w_att  = (const float*)d_in[21];
  const float* b_att  = (const float*)d_in[22];
  const float* W_ro   = (const float*)d_in[23];
  const float* W_read = (const float*)d_in[24];

  char* ws = (char*)d_ws;
  size_t off = 0;
  auto alloc = [&](size_t nfloats) {
    float* p = (float*)(ws + off);
    off += nfloats * sizeof(float);
    return p;
  };
  float* emb   = alloc((size_t)B_ * L_ * E_);        // 4 MB
  float* xgF   = alloc((size_t)B_ * L_ * G4);        // 16 MB
  float* xgB   = alloc((size_t)B_ * L_ * G4);        // 16 MB
  float* xenc  = alloc((size_t)B_ * L_ * 2 * D_);    // 8 MB
  float* xenck = alloc((size_t)B_ * L_ * D_);        // 4 MB
  float* hbufs = alloc(2 * 2 * B_ * D_);
  float* cTcat = alloc(B_ * 2 * D_);
  float* h0    = alloc(B_ * D_);
  float* c0    = alloc(B_ * D_);
  float* xin   = alloc(B_ * DI_);
  float* g     = alloc(B_ * G4);
  float* h1    = alloc(B_ * D_);
  float* c1    = alloc(B_ * D_);
  float* tq    = alloc(B_ * D_);
  float* ctx   = alloc(B_ * 2 * D_);
  float* hc    = alloc(B_ * 3 * D_);
  float* pre   = alloc(B_ * D_);
  int*   cnt   = (int*)(ws + off); off += 2 * L_ * sizeof(int);

  // 0) reset persistent-scan state (required every call / graph replay)
  init_kernel<<<128, 256, 0, stream>>>(hbufs, cnt);
  // 1) embedding gather
  embed_gather_kernel<<<B_ * L_, 128, 0, stream>>>(x_train, word_emb, emb);
  // 2) input gates: xg = emb @ Wih^T + bih   (M=2048, N=2048, K=512)
  gemm_wmma_kernel<<<dim3(128, 8), 256, 0, stream>>>(emb, E_, Wih_f, E_, bih_f, nullptr, xgF, G4, nullptr, E_, 0);
  gemm_wmma_kernel<<<dim3(128, 8), 256, 0, stream>>>(emb, E_, Wih_b, E_, bih_b, nullptr, xgB, G4, nullptr, E_, 0);
  // 3) bidirectional recurrent scan (persistent, LDS-resident weights)
  lstm_scan_kernel<<<32, 256, SCAN_SMEM_BYTES, stream>>>(
      xgF, xgB, Whh_f, Whh_b, bhh_f, bhh_b, hbufs, xenc, cTcat, cnt);
  // 4) bridge: c0 = cT_cat @ W_bridge^T ; h0 = tanh(c0)
  gemm_wmma_kernel<<<dim3(1, 2), 256, 0, stream>>>(cTcat, 2 * D_, W_bridge, 2 * D_, nullptr, nullptr, h0, D_, c0, 2 * D_, 1);
  // 5) x_enc_k = x_enc @ W_enc2k^T   (M=2048, N=512, K=1024)
  gemm_wmma_kernel<<<dim3(128, 2), 256, 0, stream>>>(xenc, 2 * D_, W_enc2k, 2 * D_, nullptr, nullptr, xenck, D_, nullptr, 2 * D_, 0);
  // 6) decoder single step
  dec_xin_kernel<<<B_, 256, 0, stream>>>(x_train, y_train, word_emb, attr_emb, xin);
  gemm_wmma_kernel<<<dim3(1, 8), 256, 0, stream>>>(xin, DI_, Wih_d, DI_, bih_d, nullptr, g, G4, nullptr, DI_, 0);
  gemm_wmma_kernel<<<dim3(1, 8), 256, 0, stream>>>(h0, D_, Whh_d, D_, bhh_d, g, g, G4, nullptr, D_, 0);
  dec_gates_kernel<<<(B_ * D_) / 256, 256, 0, stream>>>(g, c0, h1, c1);
  // 7) attention
  gemm_wmma_kernel<<<dim3(1, 2), 256, 0, stream>>>(h1, D_, W_trg, D_, b_trg, nullptr, tq, D_, nullptr, D_, 0);
  attention_kernel<<<B_, 128, 0, stream>>>(xenck, xenc, tq, w_att, b_att, ctx);
  // 8) readout
  hc_build_kernel<<<B_, 256, 0, stream>>>(h1, ctx, hc);
  gemm_wmma_kernel<<<dim3(1, 2), 256, 0, stream>>>(hc, 3 * D_, W_ro, 3 * D_, nullptr, nullptr, pre, D_, nullptr, 3 * D_, 1);
  // 9) logits = pre @ W_read^T   (M=16, N=32000, K=512) -> d_out (B, 1, V)
  gemm_wmma_kernel<<<dim3(1, 125), 256, 0, stream>>>(pre, D_, W_read, D_, nullptr, nullptr, (float*)d_out, 32000, nullptr, D_, 0);
}